// reliability_based_co_teaching_loss_26955214750390
// MI455X (gfx1250) — compile-verified
//
#include <hip/hip_runtime.h>

typedef __attribute__((ext_vector_type(2))) float v2f;
typedef __attribute__((ext_vector_type(8))) float v8f;

#define NB 4
#define NC 8
#define ND 64
#define NPIX (384 * 384)   // 147456
#define FEPS 1e-8f

// workspace layout (float offsets)
#define WS_SUMS_M 0
#define WS_SUMS_A (WS_SUMS_M + NB * NC * ND)   // 2048
#define WS_CNT_M  (WS_SUMS_A + NB * NC * ND)   // 4096
#define WS_CNT_A  (WS_CNT_M + NB * NC)         // 4128
#define WS_CEN_M  (WS_CNT_A + NB * NC)         // 4160
#define WS_CEN_A  (WS_CEN_M + NB * NC * ND)    // 6208
#define WS_NRM_M  (WS_CEN_A + NB * NC * ND)    // 8256
#define WS_NRM_A  (WS_NRM_M + NB * NC)         // 8288
#define WS_ACC    (WS_NRM_A + NB * NC)         // 8320
#define WS_TOTAL  (WS_ACC + 4)                 // 8324 floats (~33 KB)

__global__ void k_zero(float* ws) {
    int i = blockIdx.x * blockDim.x + threadIdx.x;
    if (i < WS_TOTAL) ws[i] = 0.0f;
}

__device__ __forceinline__ float readlane_f(float v, int lane) {
    return __int_as_float(__builtin_amdgcn_readlane(__float_as_int(v), lane));
}

// Pass 1: per-branch class-center accumulation via V_WMMA_F32_16X16X4_F32.
// grid = (NPIX/2048, NB), block = 256 (8 waves); each wave owns 256 pixels.
__global__ __launch_bounds__(256) void k_center_accum(
    const float* __restrict__ pred, const float* __restrict__ feat,
    float* __restrict__ sums, float* __restrict__ cnt) {
    const int b    = blockIdx.y;
    const int tid  = threadIdx.x;
    const int wave = tid >> 5;
    const int lane = tid & 31;
    const int hv   = lane >> 4;   // which half of the wave (K 0/1 vs K 2/3)
    const int m    = lane & 15;   // class row (A) / d-column (B)

    __shared__ float s_sums[NC * ND];
    __shared__ int   s_hist[NC];
    for (int i = tid; i < NC * ND; i += 256) s_sums[i] = 0.0f;
    if (tid < NC) s_hist[tid] = 0;
    __syncthreads();

    const float* pb = pred + (size_t)b * NC * NPIX;
    const float* fb = feat + (size_t)b * ND * NPIX;

    v8f acc[4];
    #pragma unroll
    for (int t = 0; t < 4; ++t)
        acc[t] = (v8f){0.f, 0.f, 0.f, 0.f, 0.f, 0.f, 0.f, 0.f};

    const int pxw = blockIdx.x * 2048 + wave * 256;

    for (int it = 0; it < 8; ++it) {
        const int pix = pxw + it * 32 + lane;

        // logits once into registers
        float lp[NC];
        #pragma unroll
        for (int c = 0; c < NC; ++c) lp[c] = pb[(size_t)c * NPIX + pix];

        // softmax confidence + argmax for this lane's pixel
        float mx = lp[0];
        int hard = 0;
        #pragma unroll
        for (int c = 1; c < NC; ++c)
            if (lp[c] > mx) { mx = lp[c]; hard = c; }
        float s = 0.0f;
        #pragma unroll
        for (int c = 0; c < NC; ++c) s += expf(lp[c] - mx);
        float conf = 1.0f / s;  // max(softmax) = exp(mx - lse) = 1/s

        atomicAdd(&s_hist[hard], 1);

        // 8 quads of 4 pixels; A = (hard==class)?conf:0 per ISA 16x4 f32 layout.
        // Fully unrolled so readlane indices are compile-time constants (SGPR broadcasts).
        #pragma unroll
        for (int q = 0; q < 8; ++q) {
            const float c0 = readlane_f(conf, 4 * q + 0);
            const float c1 = readlane_f(conf, 4 * q + 1);
            const float c2 = readlane_f(conf, 4 * q + 2);
            const float c3 = readlane_f(conf, 4 * q + 3);
            const int   h0 = __builtin_amdgcn_readlane(hard, 4 * q + 0);
            const int   h1 = __builtin_amdgcn_readlane(hard, 4 * q + 1);
            const int   h2 = __builtin_amdgcn_readlane(hard, 4 * q + 2);
            const int   h3 = __builtin_amdgcn_readlane(hard, 4 * q + 3);

            // per-half selection with constant indices only (no reg-array spills)
            const float cx = hv ? c2 : c0;
            const float cy = hv ? c3 : c1;
            const int   hx = hv ? h2 : h0;
            const int   hy = hv ? h3 : h1;

            v2f A;
            A.x = (hx == m) ? cx : 0.0f;
            A.y = (hy == m) ? cy : 0.0f;

            const int px0 = pxw + it * 32 + 4 * q + (hv ? 2 : 0);
            #pragma unroll
            for (int t = 0; t < 4; ++t) {
                const float* fp = fb + (size_t)(t * 16 + m) * NPIX;
                v2f Bv;
                Bv.x = fp[px0];
                Bv.y = fp[px0 + 1];
                acc[t] = __builtin_amdgcn_wmma_f32_16x16x4_f32(
                    false, A, false, Bv, (short)0, acc[t], false, false);
            }
        }
    }

    // C/D layout: VGPR v, lanes 0-15 hold (class=v, d=16t+lane); classes 8..15 are zero
    if (lane < 16) {
        #pragma unroll
        for (int t = 0; t < 4; ++t) {
            #pragma unroll
            for (int v = 0; v < NC; ++v)
                atomicAdd(&s_sums[v * ND + t * 16 + lane], acc[t][v]);
        }
    }
    __syncthreads();
    for (int i = tid; i < NC * ND; i += 256)
        atomicAdd(&sums[b * NC * ND + i], s_sums[i]);
    if (tid < NC) atomicAdd(&cnt[b * NC + tid], (float)s_hist[tid]);
}

// Pass 2: centers = sums / cnt ; per-class center norms. Single block.
__global__ __launch_bounds__(256) void k_finalize(float* ws) {
    const int tid = threadIdx.x;
    float* sums_m = ws + WS_SUMS_M;
    float* sums_a = ws + WS_SUMS_A;
    float* cnt_m  = ws + WS_CNT_M;
    float* cnt_a  = ws + WS_CNT_A;
    float* cen_m  = ws + WS_CEN_M;
    float* cen_a  = ws + WS_CEN_A;
    float* nrm_m  = ws + WS_NRM_M;
    float* nrm_a  = ws + WS_NRM_A;

    for (int i = tid; i < NB * NC * ND; i += 256) {
        int bc = i >> 6;  // b*NC + c
        cen_m[i] = sums_m[i] / cnt_m[bc];
        cen_a[i] = sums_a[i] / cnt_a[bc];
    }
    __syncthreads();
    if (tid < NB * NC) {
        float sm = 0.0f, sa = 0.0f;
        for (int d = 0; d < ND; ++d) {
            float x = cen_m[tid * ND + d]; sm += x * x;
            float y = cen_a[tid * ND + d]; sa += y * y;
        }
        nrm_m[tid] = sqrtf(sm);
        nrm_a[tid] = sqrtf(sa);
    }
}

// Pass 3: per-pixel reliability + CE, reduce 4 scalars. grid=(NPIX/256, NB).
__global__ __launch_bounds__(256) void k_loss(
    const float* __restrict__ pred_m, const float* __restrict__ pred_a,
    const float* __restrict__ feat_m, const float* __restrict__ feat_a,
    const float* __restrict__ cen_m, const float* __restrict__ cen_a,
    const float* __restrict__ nrm_m, const float* __restrict__ nrm_a,
    float* __restrict__ accum) {
    const int b   = blockIdx.y;
    const int tid = threadIdx.x;
    const int n   = blockIdx.x * 256 + tid;

    __shared__ float s_cm[NC * ND], s_ca[NC * ND], s_nm[NC], s_na[NC];
    for (int i = tid; i < NC * ND; i += 256) {
        s_cm[i] = cen_m[b * NC * ND + i];
        s_ca[i] = cen_a[b * NC * ND + i];
    }
    if (tid < NC) { s_nm[tid] = nrm_m[b * NC + tid]; s_na[tid] = nrm_a[b * NC + tid]; }
    __syncthreads();

    const float* pm = pred_m + (size_t)b * NC * NPIX;
    const float* pa = pred_a + (size_t)b * NC * NPIX;
    const float* fm = feat_m + (size_t)b * ND * NPIX;
    const float* fa = feat_a + (size_t)b * ND * NPIX;

    float lm[NC], la[NC];
    #pragma unroll
    for (int c = 0; c < NC; ++c) {
        lm[c] = pm[(size_t)c * NPIX + n];
        la[c] = pa[(size_t)c * NPIX + n];
    }

    float mxm = lm[0]; int hm = 0;
    float mxa = la[0]; int ha = 0;
    #pragma unroll
    for (int c = 1; c < NC; ++c) {
        if (lm[c] > mxm) { mxm = lm[c]; hm = c; }
        if (la[c] > mxa) { mxa = la[c]; ha = c; }
    }
    float sm = 0.0f, sa = 0.0f;
    #pragma unroll
    for (int c = 0; c < NC; ++c) { sm += expf(lm[c] - mxm); sa += expf(la[c] - mxa); }
    const float lsem = mxm + logf(sm), lsea = mxa + logf(sa);
    const float confm = 1.0f / sm, confa = 1.0f / sa;

    const int pseudo = (confm >= confa) ? hm : ha;
    float pm_sel = 0.0f, pa_sel = 0.0f;
    #pragma unroll
    for (int c = 0; c < NC; ++c) {
        pm_sel = (c == pseudo) ? lm[c] : pm_sel;
        pa_sel = (c == pseudo) ? la[c] : pa_sel;
    }
    const float ce_m = lsem - pm_sel;
    const float ce_a = lsea - pa_sel;

    float dm = 0.0f, qm = 0.0f, da = 0.0f, qa = 0.0f;
    const float* cmrow = &s_cm[hm * ND];
    const float* carow = &s_ca[ha * ND];
    #pragma unroll 4
    for (int d = 0; d < ND; ++d) {
        float f1 = fm[(size_t)d * NPIX + n]; dm += f1 * cmrow[d]; qm += f1 * f1;
        float f2 = fa[(size_t)d * NPIX + n]; da += f2 * carow[d]; qa += f2 * f2;
    }
    const float rel_m = dm / (fmaxf(sqrtf(qm), FEPS) * fmaxf(s_nm[hm], FEPS));
    const float rel_a = da / (fmaxf(sqrtf(qa), FEPS) * fmaxf(s_na[ha], FEPS));

    float v0 = ce_m * rel_a, v1 = rel_a, v2 = ce_a * rel_m, v3 = rel_m;
    #pragma unroll
    for (int off = 16; off > 0; off >>= 1) {
        v0 += __shfl_down(v0, off, 32);
        v1 += __shfl_down(v1, off, 32);
        v2 += __shfl_down(v2, off, 32);
        v3 += __shfl_down(v3, off, 32);
    }
    if ((tid & 31) == 0) {
        atomicAdd(&accum[0], v0);
        atomicAdd(&accum[1], v1);
        atomicAdd(&accum[2], v2);
        atomicAdd(&accum[3], v3);
    }
}

__global__ void k_final(const float* __restrict__ accum, float* __restrict__ out) {
    if (threadIdx.x == 0 && blockIdx.x == 0)
        out[0] = accum[0] / accum[1] + accum[2] / accum[3];
}

extern "C" void kernel_launch(void* const* d_in, const int* in_sizes, int n_in,
                              void* d_out, int out_size, void* d_ws, size_t ws_size,
                              hipStream_t stream) {
    const float* pred_m = (const float*)d_in[0];
    const float* pred_a = (const float*)d_in[1];
    const float* feat_m = (const float*)d_in[2];
    const float* feat_a = (const float*)d_in[3];
    float* ws  = (float*)d_ws;
    float* out = (float*)d_out;

    k_zero<<<dim3((WS_TOTAL + 255) / 256), dim3(256), 0, stream>>>(ws);

    dim3 gA(NPIX / 2048, NB);
    k_center_accum<<<gA, dim3(256), 0, stream>>>(pred_m, feat_m, ws + WS_SUMS_M, ws + WS_CNT_M);
    k_center_accum<<<gA, dim3(256), 0, stream>>>(pred_a, feat_a, ws + WS_SUMS_A, ws + WS_CNT_A);

    k_finalize<<<dim3(1), dim3(256), 0, stream>>>(ws);

    dim3 gC(NPIX / 256, NB);
    k_loss<<<gC, dim3(256), 0, stream>>>(pred_m, pred_a, feat_m, feat_a,
                                         ws + WS_CEN_M, ws + WS_CEN_A,
                                         ws + WS_NRM_M, ws + WS_NRM_A,
                                         ws + WS_ACC);

    k_final<<<dim3(1), dim3(1), 0, stream>>>(ws + WS_ACC, out);
}